// masked_region_MSE_Loss_75642964017456
// MI455X (gfx1250) — compile-verified
//
#include <hip/hip_runtime.h>

// masked_region_MSE_Loss for MI455X (gfx1250, wave32)
// x,y: [B=8, C=2, H=256, W=4096] fp32; per-batch mask window of width w (=1024).
// Bandwidth-bound: read only the in-window columns (32 MB instead of 128 MB).
// Interior of the window is read with unconditional float4 (global_load_b128);
// the <=3 ragged elements at each end are handled by a few scalar lanes.

typedef __attribute__((ext_vector_type(2))) float v2f;
typedef __attribute__((ext_vector_type(8))) float v8f;

#define BB 8
#define CC 2
#define HH 256
#define WW 4096
#define NROWS (BB * CC * HH)   // 4096 rows of length W
#define NTHREADS 256

// ds_swizzle XOR lane swap within wave32 (offset encoding: xor<<10 | and=0x1f)
__device__ __forceinline__ float swz_xor1(float v) {
    return __builtin_bit_cast(float,
        __builtin_amdgcn_ds_swizzle(__builtin_bit_cast(int, v), 0x041F));
}
__device__ __forceinline__ float swz_xor2(float v) {
    return __builtin_bit_cast(float,
        __builtin_amdgcn_ds_swizzle(__builtin_bit_cast(int, v), 0x081F));
}
__device__ __forceinline__ float swz_xor4(float v) {
    return __builtin_bit_cast(float,
        __builtin_amdgcn_ds_swizzle(__builtin_bit_cast(int, v), 0x101F));
}
__device__ __forceinline__ float swz_xor8(float v) {
    return __builtin_bit_cast(float,
        __builtin_amdgcn_ds_swizzle(__builtin_bit_cast(int, v), 0x201F));
}

// ---------------------------------------------------------------------------
// Kernel 1: one block per (b,c,h) row.
// ---------------------------------------------------------------------------
__global__ __launch_bounds__(NTHREADS)
void masked_mse_rows(const float* __restrict__ x,
                     const float* __restrict__ y,
                     const int*  __restrict__ mask_pos,   // [B]
                     const int*  __restrict__ mask_width, // [1]
                     float*      __restrict__ partial)    // [NROWS]
{
    const int row = blockIdx.x;          // row = ((b*C)+c)*H + h
    const int tid = threadIdx.x;
    const int b   = row / (CC * HH);

    const int w  = mask_width[0];
    int s = mask_pos[b];
    s = min(max(s, 0), WW - w);          // clamp like reference
    const int e    = s + w;              // exclusive end
    const int c_lo = (s + 3) & ~3;       // first 16B-aligned element >= s
    const int c_hi = e & ~3;             // last 16B-aligned boundary <= e
    const int n4   = (c_hi - c_lo) >> 2; // full float4 chunks (255..256)

    const size_t rbase = (size_t)row * WW;
    const float4* __restrict__ xr = (const float4*)(x + rbase + c_lo);
    const float4* __restrict__ yr = (const float4*)(y + rbase + c_lo);

    float acc = 0.0f;
    // Unmasked interior: pure b128 streaming loads.
    for (int i = tid; i < n4; i += NTHREADS) {
        const float4 xv = xr[i];
        const float4 yv = yr[i];
        const float d0 = xv.x - yv.x;
        const float d1 = xv.y - yv.y;
        const float d2 = xv.z - yv.z;
        const float d3 = xv.w - yv.w;
        acc = fmaf(d0, d0, acc);
        acc = fmaf(d1, d1, acc);
        acc = fmaf(d2, d2, acc);
        acc = fmaf(d3, d3, acc);
    }

    // Ragged edges: <=3 leading + <=3 trailing scalar elements.
    const int nlead = c_lo - s;          // 0..3
    const int ntail = e - c_hi;          // 0..3
    const int ne    = nlead + ntail;     // 0..6
    if (tid < ne) {
        const int col = (tid < nlead) ? (s + tid) : (c_hi + (tid - nlead));
        const float d = x[rbase + col] - y[rbase + col];
        acc = fmaf(d, d, acc);
    }

    // ---- wave32 reduction via the FP32 matrix engine -----------------------
    // A = ones(16x4); B VGPR0 carries the 32 lane partials (two K-rows,
    // n = lane % 16), B VGPR1 = 0. D[m][n] = acc[n] + acc[n+16] for every m.
    // EXEC is all ones here (block fully active and reconverged).
    v2f a;  a[0] = 1.0f; a[1] = 1.0f;
    v2f bm; bm[0] = acc; bm[1] = 0.0f;
    v8f c = {};
    c = __builtin_amdgcn_wmma_f32_16x16x4_f32(
            /*neg_a=*/false, a, /*neg_b=*/false, bm,
            /*c_mod=*/(short)0, c, /*reuse_a=*/false, /*reuse_b=*/false);

    // c[0]: lanes 0..15 hold the 16 column sums (lanes 16..31 hold the same
    // values from row M=8). XOR-swizzle butterfly sums them within each half.
    float t = c[0];
    t += swz_xor1(t);
    t += swz_xor2(t);
    t += swz_xor4(t);
    t += swz_xor8(t);   // every lane now holds the full wave-of-32 sum

    __shared__ float lds[NTHREADS / 32];
    const int wave = tid >> 5;
    if ((tid & 31) == 0) lds[wave] = t;
    __syncthreads();
    if (tid == 0) {
        float bs = 0.0f;
#pragma unroll
        for (int i = 0; i < NTHREADS / 32; ++i) bs += lds[i];
        partial[row] = bs;
    }
}

// ---------------------------------------------------------------------------
// Kernel 2: deterministic final reduce of the 4096 row partials + mean scale.
// ---------------------------------------------------------------------------
__global__ __launch_bounds__(NTHREADS)
void masked_mse_finalize(const float* __restrict__ partial,
                         float* __restrict__ out)
{
    __shared__ float lds[NTHREADS];
    float a = 0.0f;
    for (int i = threadIdx.x; i < NROWS; i += NTHREADS) a += partial[i];
    lds[threadIdx.x] = a;
    __syncthreads();
#pragma unroll
    for (int off = NTHREADS / 2; off > 0; off >>= 1) {
        if (threadIdx.x < off) lds[threadIdx.x] += lds[threadIdx.x + off];
        __syncthreads();
    }
    if (threadIdx.x == 0) {
        const float inv_n =
            1.0f / ((float)BB * (float)CC * (float)HH * (float)WW);
        out[0] = lds[0] * inv_n;
    }
}

extern "C" void kernel_launch(void* const* d_in, const int* in_sizes, int n_in,
                              void* d_out, int out_size, void* d_ws, size_t ws_size,
                              hipStream_t stream) {
    const float* x          = (const float*)d_in[0];
    const float* y          = (const float*)d_in[1];
    const int*   mask_pos   = (const int*)d_in[2];
    const int*   mask_width = (const int*)d_in[3];
    float*       out        = (float*)d_out;
    float*       partial    = (float*)d_ws;   // NROWS * 4 = 16 KB scratch

    masked_mse_rows<<<NROWS, NTHREADS, 0, stream>>>(x, y, mask_pos, mask_width,
                                                    partial);
    masked_mse_finalize<<<1, NTHREADS, 0, stream>>>(partial, out);
}